// CommonsenseRNNCell_59605556133901
// MI455X (gfx1250) — compile-verified
//
#include <hip/hip_runtime.h>
#include <hip/hip_bf16.h>
#include <cmath>

// Problem constants (from reference): B=4096, P=2, T=64, DR=1024, DC=768, G=256
#define Bsz 4096
#define Pp  2
#define Tt  64
#define Gg  256

typedef float v2f __attribute__((ext_vector_type(2)));
typedef float v8f __attribute__((ext_vector_type(8)));

// ---------------------------------------------------------------------------
// Segment descriptor: gates[n,m] += sum_k X[rowmap(n),k] * W[m,k]
//   rowmap(n) = n >> rowShift   (rowShift=1 implements broadcast over P=2)
//   W points at a column slice of Wih (or at Whh), ldw = full row length.
// ---------------------------------------------------------------------------
struct Seg {
  const float* X;
  const float* W;
  int ldx, ldw, K, rowShift;
};
struct GemmArgs {
  Seg   seg[5];
  int   nseg, N;
  float* out;   // N x 1024, row-major
};

// ---------------------------------------------------------------------------
// WMMA f32 GEMM: block = 64(N) x 64(M) tile, 256 threads = 8 waves.
// Wave w: n-subtile = w%4, m-subtiles {2*(w/4), 2*(w/4)+1}.
// K staged through LDS in chunks of 32; V_WMMA_F32_16X16X4_F32 inner loop.
// A-layout (ISA 7.12.2, 32-bit 16x4): lane L (row = L%16) holds K = kb, kb+1
// with kb = k0 + 2*(L/16).  B mirrors with N = L%16 columns of W.
// C/D: VGPR i, lane L -> [M = i + 8*(L/16), N = L%16].
// ---------------------------------------------------------------------------
__global__ __launch_bounds__(256) void gemm_gates(GemmArgs ga) {
  __shared__ float As[64][36];
  __shared__ float Bs[64][36];
  const int tid  = threadIdx.x;
  const int wave = tid >> 5;
  const int lane = tid & 31;
  const int n0 = blockIdx.x * 64;
  const int m0 = blockIdx.y * 64;
  const int nsub = wave & 3;
  const int msub = (wave >> 2) * 2;

  v8f acc0 = {0.f,0.f,0.f,0.f,0.f,0.f,0.f,0.f};
  v8f acc1 = {0.f,0.f,0.f,0.f,0.f,0.f,0.f,0.f};

  const int lrow = tid >> 3;        // 0..31
  const int lcol = (tid & 7) << 2;  // 0,4,...,28

  for (int s = 0; s < ga.nseg; ++s) {
    const float* __restrict__ X = ga.seg[s].X;
    const float* __restrict__ W = ga.seg[s].W;
    const int ldx = ga.seg[s].ldx;
    const int ldw = ga.seg[s].ldw;
    const int K   = ga.seg[s].K;      // multiple of 32 for all cells here
    const int rsh = ga.seg[s].rowShift;
    for (int k0 = 0; k0 < K; k0 += 32) {
      #pragma unroll
      for (int h = 0; h < 2; ++h) {
        const int row = lrow + 32*h;
        const int gr  = (n0 + row) >> rsh;
        const float4 va = *(const float4*)(X + (size_t)gr*ldx + (k0 + lcol));
        As[row][lcol+0]=va.x; As[row][lcol+1]=va.y;
        As[row][lcol+2]=va.z; As[row][lcol+3]=va.w;
        const float4 vb = *(const float4*)(W + (size_t)(m0+row)*ldw + (k0 + lcol));
        Bs[row][lcol+0]=vb.x; Bs[row][lcol+1]=vb.y;
        Bs[row][lcol+2]=vb.z; Bs[row][lcol+3]=vb.w;
      }
      __syncthreads();
      const int r  = lane & 15;
      const int kh = (lane >> 4) << 1;
      #pragma unroll
      for (int kk = 0; kk < 32; kk += 4) {
        const int kb = kk + kh;
        v2f a, b0, b1;
        a.x  = As[nsub*16 + r][kb];      a.y  = As[nsub*16 + r][kb+1];
        b0.x = Bs[msub*16 + r][kb];      b0.y = Bs[msub*16 + r][kb+1];
        b1.x = Bs[(msub+1)*16 + r][kb];  b1.y = Bs[(msub+1)*16 + r][kb+1];
        acc0 = __builtin_amdgcn_wmma_f32_16x16x4_f32(false, a, false, b0,
                                                     (short)0, acc0, false, false);
        acc1 = __builtin_amdgcn_wmma_f32_16x16x4_f32(false, a, false, b1,
                                                     (short)0, acc1, false, false);
      }
      __syncthreads();
    }
  }
  const int r   = lane & 15;
  const int mup = (lane >> 4) * 8;
  float* __restrict__ out = ga.out;
  #pragma unroll
  for (int i = 0; i < 8; ++i) {
    const int row = n0 + nsub*16 + i + mup;
    const int col = m0 + msub*16 + r;
    out[(size_t)row*1024 + col]      = acc0[i];
    out[(size_t)row*1024 + col + 16] = acc1[i];
  }
}

// ---------------------------------------------------------------------------
// LSTM pointwise: gates(N,1024) + bias -> h2,c2 (N,256). Gate order i,f,g,o.
// ---------------------------------------------------------------------------
__device__ __forceinline__ float sigm(float x) { return 1.0f / (1.0f + expf(-x)); }

__global__ __launch_bounds__(256) void lstm_pointwise(
    const float* __restrict__ gates, const float* __restrict__ bih,
    const float* __restrict__ bhh,   const float* __restrict__ cprev,
    float* __restrict__ hout, float* __restrict__ cout) {
  const int idx = blockIdx.x * 256 + threadIdx.x;   // < N*256 (grid = N)
  const int n = idx >> 8;
  const int g = idx & 255;
  const float* gr = gates + (size_t)n * 1024;
  const float gi = gr[g]       + bih[g]       + bhh[g];
  const float gf = gr[256 + g] + bih[256 + g] + bhh[256 + g];
  const float gg = gr[512 + g] + bih[512 + g] + bhh[512 + g];
  const float go = gr[768 + g] + bih[768 + g] + bhh[768 + g];
  const float c2 = sigm(gf) * cprev[idx] + sigm(gi) * tanhf(gg);
  hout[idx] = sigm(go) * tanhf(c2);
  cout[idx] = c2;
}

// ---------------------------------------------------------------------------
// qmask argmax (P=2, first-max wins ties like jnp.argmax)
// ---------------------------------------------------------------------------
__global__ void qargmax(const float* __restrict__ qmask, int* __restrict__ qidx) {
  const int b = blockIdx.x * 256 + threadIdx.x;
  if (b < Bsz) qidx[b] = (qmask[2*b + 1] > qmask[2*b]) ? 1 : 0;
}

// sel: dst[b,g] = src[b, qidx[b], g]   src is (B,P,G)
__global__ void gather_sel(const float* __restrict__ src,
                           const int* __restrict__ qidx,
                           float* __restrict__ dst) {
  const int idx = blockIdx.x * 256 + threadIdx.x;   // < B*G
  const int b = idx >> 8;
  const int g = idx & 255;
  dst[idx] = src[(((size_t)b * Pp + qidx[b]) << 8) + g];
}

// out = a*(1-qm) + b*qm for both h and c streams; qm = qmask[b,p]
__global__ void mix2(const float* __restrict__ ah, const float* __restrict__ bh,
                     float* __restrict__ oh,
                     const float* __restrict__ ac, const float* __restrict__ bc,
                     float* __restrict__ oc,
                     const float* __restrict__ qmask) {
  const int idx = blockIdx.x * 256 + threadIdx.x;   // < B*P*G
  const int bp = idx >> 8;                          // b*P + p
  const float qm = qmask[bp];
  oh[idx] = ah[idx] * (1.0f - qm) + bh[idx] * qm;
  oc[idx] = ac[idx] * (1.0f - qm) + bc[idx] * qm;
}

// ---------------------------------------------------------------------------
// Attention: scale[t] = <g_hist[t,b,:], attn_w>; softmax over t;
// c_[b,:] = sum_t alpha[t] * g_hist[t,b,:]; alpha_out[b,t] = alpha[t].
// One block (256 threads) per batch element.
// ---------------------------------------------------------------------------
__global__ __launch_bounds__(256) void attn_kernel(
    const float* __restrict__ g_hist, const float* __restrict__ attn_w,
    float* __restrict__ c_out, float* __restrict__ alpha_out) {
  const int b = blockIdx.x;
  const int tid = threadIdx.x;
  __shared__ float part[Tt][4];
  __shared__ float sc[Tt];
  __shared__ float al[Tt];

  const int t  = tid >> 2;
  const int qq = tid & 3;
  {
    const float* gh = g_hist + ((size_t)t * Bsz + b) * Gg + qq * 64;
    const float* aw = attn_w + qq * 64;
    float s = 0.f;
    #pragma unroll 8
    for (int g = 0; g < 64; ++g) s += gh[g] * aw[g];
    part[t][qq] = s;
  }
  __syncthreads();
  if (tid < Tt) sc[tid] = part[tid][0] + part[tid][1] + part[tid][2] + part[tid][3];
  __syncthreads();
  if (tid == 0) {
    float m = sc[0];
    for (int i = 1; i < Tt; ++i) m = fmaxf(m, sc[i]);
    float sum = 0.f;
    for (int i = 0; i < Tt; ++i) { float e = expf(sc[i] - m); al[i] = e; sum += e; }
    const float inv = 1.f / sum;
    for (int i = 0; i < Tt; ++i) al[i] *= inv;
  }
  __syncthreads();
  float acc = 0.f;
  for (int i = 0; i < Tt; ++i)
    acc += al[i] * g_hist[((size_t)i * Bsz + b) * Gg + tid];
  c_out[(size_t)b * Gg + tid] = acc;
  if (tid < Tt) alpha_out[(size_t)b * Tt + tid] = al[tid];
}

// ---------------------------------------------------------------------------
static inline Seg mkseg(const float* X, int ldx, int rsh,
                        const float* W, int ldw, int K) {
  Seg s; s.X = X; s.W = W; s.ldx = ldx; s.ldw = ldw; s.K = K; s.rowShift = rsh;
  return s;
}

extern "C" void kernel_launch(void* const* d_in, const int* in_sizes, int n_in,
                              void* d_out, int out_size, void* d_ws, size_t ws_size,
                              hipStream_t stream) {
  (void)in_sizes; (void)n_in; (void)out_size; (void)ws_size;
  const float* U      = (const float*)d_in[0];
  const float* x1     = (const float*)d_in[1];
  const float* x2     = (const float*)d_in[2];
  const float* x3     = (const float*)d_in[3];
  const float* o1     = (const float*)d_in[4];
  const float* o2     = (const float*)d_in[5];
  const float* qmask  = (const float*)d_in[6];
  const float* g_hist = (const float*)d_in[7];
  const float* gc_hist= (const float*)d_in[8];
  const float* q0     = (const float*)d_in[9];
  const float* qc0    = (const float*)d_in[10];
  const float* r0     = (const float*)d_in[11];
  const float* rc0    = (const float*)d_in[12];
  const float* i0     = (const float*)d_in[13];
  const float* ic0    = (const float*)d_in[14];
  const float* e0     = (const float*)d_in[15];
  const float* ec0    = (const float*)d_in[16];
  // param tuples flatten as (Wih, Whh, bih, bhh)
  const float* Wg  = (const float*)d_in[17]; const float* Hg  = (const float*)d_in[18];
  const float* big = (const float*)d_in[19]; const float* bhg = (const float*)d_in[20];
  const float* Wp  = (const float*)d_in[21]; const float* Hp  = (const float*)d_in[22];
  const float* bip = (const float*)d_in[23]; const float* bhp = (const float*)d_in[24];
  const float* Wr  = (const float*)d_in[25]; const float* Hr  = (const float*)d_in[26];
  const float* bir = (const float*)d_in[27]; const float* bhr = (const float*)d_in[28];
  const float* Wi  = (const float*)d_in[29]; const float* Hi  = (const float*)d_in[30];
  const float* bii = (const float*)d_in[31]; const float* bhi = (const float*)d_in[32];
  const float* We  = (const float*)d_in[33]; const float* He  = (const float*)d_in[34];
  const float* bie = (const float*)d_in[35]; const float* bhe = (const float*)d_in[36];
  const float* Wpl = (const float*)d_in[37]; const float* Hpl = (const float*)d_in[38];
  const float* bipl= (const float*)d_in[39]; const float* bhpl= (const float*)d_in[40];
  const float* Wrl = (const float*)d_in[41]; const float* Hrl = (const float*)d_in[42];
  const float* birl= (const float*)d_in[43]; const float* bhrl= (const float*)d_in[44];
  const float* attn_w = (const float*)d_in[45];

  float* out = (float*)d_out;
  float* o_g  = out + 0;
  float* o_q  = out + 1048576;
  float* o_r  = out + 3145728;
  float* o_i  = out + 5242880;
  float* o_e  = out + 7340032;
  float* o_gc = out + 8388608;
  float* o_qc = out + 9437184;
  float* o_rc = out + 11534336;
  float* o_ic = out + 13631488;
  float* o_ec = out + 15728640;
  float* o_al = out + 16777216;

  float* wsf = (float*)d_ws;
  int*   qidx = (int*)d_ws;                 // B ints
  float* c_    = wsf + 4096;                // B*256
  float* sel0  = wsf + 1052672;             // B*256
  float* sel1  = wsf + 2101248;             // B*256
  float* sel2  = wsf + 3149824;             // B*256
  float* buf0  = wsf + 4198400;             // B*P*256 each
  float* buf1  = wsf + 6295552;
  float* buf2  = wsf + 8392704;
  float* buf3  = wsf + 10489856;
  float* buf4  = wsf + 12587008;
  float* buf5  = wsf + 14684160;
  float* gates = wsf + 16781312;            // 8192*1024

  const float* gh_last  = g_hist  + (size_t)(Tt - 1) * Bsz * Gg;
  const float* ghc_last = gc_hist + (size_t)(Tt - 1) * Bsz * Gg;

  const dim3 gemmB(256), gemmG4(64, 16), gemmG8(128, 16);

  // ---- preliminaries -------------------------------------------------------
  qargmax<<<16, 256, 0, stream>>>(qmask, qidx);
  attn_kernel<<<Bsz, 256, 0, stream>>>(g_hist, attn_w, c_, o_al);
  gather_sel<<<4096, 256, 0, stream>>>(q0, qidx, sel0);
  gather_sel<<<4096, 256, 0, stream>>>(r0, qidx, sel1);

  // ---- g cell (N=B) --------------------------------------------------------
  {
    GemmArgs ga{}; ga.nseg = 4; ga.N = Bsz; ga.out = gates;
    ga.seg[0] = mkseg(U,       1024, 0, Wg,        1536, 1024);
    ga.seg[1] = mkseg(sel0,     256, 0, Wg + 1024, 1536,  256);
    ga.seg[2] = mkseg(sel1,     256, 0, Wg + 1280, 1536,  256);
    ga.seg[3] = mkseg(gh_last,  256, 0, Hg,         256,  256);
    gemm_gates<<<gemmG4, gemmB, 0, stream>>>(ga);
    lstm_pointwise<<<Bsz, 256, 0, stream>>>(gates, big, bhg, ghc_last, o_g, o_gc);
  }
  // ---- rs (r_params, N=B*P) -----------------------------------------------
  {
    GemmArgs ga{}; ga.nseg = 4; ga.N = Bsz * Pp; ga.out = gates;
    ga.seg[0] = mkseg(U,   1024, 1, Wr,        2048, 1024);
    ga.seg[1] = mkseg(x2,   768, 1, Wr + 1024, 2048,  768);
    ga.seg[2] = mkseg(c_,   256, 1, Wr + 1792, 2048,  256);
    ga.seg[3] = mkseg(r0,   256, 0, Hr,         256,  256);
    gemm_gates<<<gemmG8, gemmB, 0, stream>>>(ga);
    lstm_pointwise<<<Bsz * Pp, 256, 0, stream>>>(gates, bir, bhr, rc0, buf0, buf1);
  }
  gather_sel<<<4096, 256, 0, stream>>>(buf0, qidx, sel0);   // sel(rs_)
  // ---- qs (p_params, N=B*P) -----------------------------------------------
  {
    GemmArgs ga{}; ga.nseg = 3; ga.N = Bsz * Pp; ga.out = gates;
    ga.seg[0] = mkseg(x1,  768, 1, Wp,       1024,  768);
    ga.seg[1] = mkseg(c_,  256, 1, Wp + 768, 1024,  256);
    ga.seg[2] = mkseg(q0,  256, 0, Hp,        256,  256);
    gemm_gates<<<gemmG8, gemmB, 0, stream>>>(ga);
    lstm_pointwise<<<Bsz * Pp, 256, 0, stream>>>(gates, bip, bhp, qc0, buf2, buf3);
  }
  gather_sel<<<4096, 256, 0, stream>>>(buf2, qidx, sel1);   // sel(qs_)
  // ---- rl (rl_params) ------------------------------------------------------
  {
    GemmArgs ga{}; ga.nseg = 4; ga.N = Bsz * Pp; ga.out = gates;
    ga.seg[0] = mkseg(U,    1024, 1, Wrl,        2048, 1024);
    ga.seg[1] = mkseg(o2,    768, 1, Wrl + 1024, 2048,  768);
    ga.seg[2] = mkseg(sel0,  256, 1, Wrl + 1792, 2048,  256);
    ga.seg[3] = mkseg(r0,    256, 0, Hrl,         256,  256);
    gemm_gates<<<gemmG8, gemmB, 0, stream>>>(ga);
    lstm_pointwise<<<Bsz * Pp, 256, 0, stream>>>(gates, birl, bhrl, rc0, buf4, buf5);
  }
  mix2<<<8192, 256, 0, stream>>>(buf4, buf0, o_r, buf5, buf1, o_rc, qmask);
  // ---- ql (pl_params) ------------------------------------------------------
  {
    GemmArgs ga{}; ga.nseg = 3; ga.N = Bsz * Pp; ga.out = gates;
    ga.seg[0] = mkseg(o1,    768, 1, Wpl,       1024,  768);
    ga.seg[1] = mkseg(sel1,  256, 1, Wpl + 768, 1024,  256);
    ga.seg[2] = mkseg(q0,    256, 0, Hpl,        256,  256);
    gemm_gates<<<gemmG8, gemmB, 0, stream>>>(ga);
    lstm_pointwise<<<Bsz * Pp, 256, 0, stream>>>(gates, bipl, bhpl, qc0, buf0, buf1);
  }
  mix2<<<8192, 256, 0, stream>>>(buf0, buf2, o_q, buf1, buf3, o_qc, qmask);
  gather_sel<<<4096, 256, 0, stream>>>(o_q, qidx, sel0);    // sel(q_)
  // ---- is (i_params) -------------------------------------------------------
  {
    GemmArgs ga{}; ga.nseg = 3; ga.N = Bsz * Pp; ga.out = gates;
    ga.seg[0] = mkseg(x3,    768, 1, Wi,       1024,  768);
    ga.seg[1] = mkseg(sel0,  256, 1, Wi + 768, 1024,  256);
    ga.seg[2] = mkseg(i0,    256, 0, Hi,        256,  256);
    gemm_gates<<<gemmG8, gemmB, 0, stream>>>(ga);
    lstm_pointwise<<<Bsz * Pp, 256, 0, stream>>>(gates, bii, bhi, ic0, buf2, buf3);
  }
  mix2<<<8192, 256, 0, stream>>>(i0, buf2, o_i, ic0, buf3, o_ic, qmask);
  gather_sel<<<4096, 256, 0, stream>>>(o_r, qidx, sel1);    // sel(r_)
  gather_sel<<<4096, 256, 0, stream>>>(o_i, qidx, sel2);    // sel(i_)
  // ---- e cell (N=B) --------------------------------------------------------
  {
    GemmArgs ga{}; ga.nseg = 5; ga.N = Bsz; ga.out = gates;
    ga.seg[0] = mkseg(U,     1024, 0, We,        1792, 1024);
    ga.seg[1] = mkseg(sel0,   256, 0, We + 1024, 1792,  256);
    ga.seg[2] = mkseg(sel1,   256, 0, We + 1280, 1792,  256);
    ga.seg[3] = mkseg(sel2,   256, 0, We + 1536, 1792,  256);
    ga.seg[4] = mkseg(e0,     256, 0, He,         256,  256);
    gemm_gates<<<gemmG4, gemmB, 0, stream>>>(ga);
    lstm_pointwise<<<Bsz, 256, 0, stream>>>(gates, bie, bhe, ec0, o_e, o_ec);
  }
}